// STGraphConstructor_7172595384462
// MI455X (gfx1250) — compile-verified
//
#include <hip/hip_runtime.h>

typedef __attribute__((ext_vector_type(2))) float v2f;
typedef __attribute__((ext_vector_type(4))) float v4f;
typedef __attribute__((ext_vector_type(8))) float v8f;
typedef __attribute__((ext_vector_type(4))) unsigned int u32x4;
typedef __attribute__((ext_vector_type(4))) int i32x4;
typedef __attribute__((ext_vector_type(8))) int i32x8;

#define NSPAT 2048
#define NTEMP 64
#define MTOT  2112
#define DDIM  64
#define LDS_STRIDE 68   // 64 + 4-float pad => 4-bank row skew, conflict-free ds_load_b64

// ---------------- CDNA5 v_tanh_f32 (single TRANS instruction) ----------------
__device__ __forceinline__ float fast_tanh(float x) {
#if __has_builtin(__builtin_amdgcn_tanhf)
  return __builtin_amdgcn_tanhf(x);
#else
  float r;
  asm("v_tanh_f32 %0, %1" : "=v"(r) : "v"(x));
  return r;
#endif
}

// ---------------- TDM: DMA one 64x64 f32 tile (row stride 64) into LDS -------
// LDS destination gets 4 DWORDs of padding after every 64 DWORDs (row), i.e.
// hardware reproduces the 68-float LDS row stride used by the WMMA frag reads.
#if __has_builtin(__builtin_amdgcn_tensor_load_to_lds)
#define USE_TDM 1
__device__ __forceinline__ void tdm_load_tile_64x64(const float* gsrc, float* ldsDst) {
  const unsigned long long ga = (unsigned long long)(size_t)gsrc;
  const unsigned lds_addr = (unsigned)(size_t)ldsDst;   // low 32 bits = LDS byte addr

  u32x4 g0;
  g0[0] = 1u;                                              // count=1, user descriptor
  g0[1] = lds_addr;                                        // bits 63:32
  g0[2] = (unsigned)(ga & 0xffffffffu);                    // global_addr[31:0]
  g0[3] = ((unsigned)(ga >> 32) & 0x01ffffffu)             // global_addr[56:32]
          | (2u << 30);                                    // type=2 ("image")

  i32x8 g1;
  g1[0] = (int)((2u << 16)      // data_size = 4 bytes
              | (1u << 20)      // pad_enable
              | (5u << 22)      // pad_interval: 64 DWORDs
              | (3u << 25));    // pad_amount:   4 DWORDs
  g1[1] = (int)(64u << 16);     // tensor_dim0 = 64        (bits 79:48)
  g1[2] = (int)(64u << 16);     // tensor_dim1 = 64        (bits 111:80)
  g1[3] = (int)(64u << 16);     // tile_dim0   = 64        (bits 127:112)
  g1[4] = 64;                   // tile_dim1 = 64, tile_dim2 = 0
  g1[5] = 64;                   // tensor_dim0_stride = 64 elements (48-bit lo)
  g1[6] = 0;                    // stride hi / dim1_stride lo
  g1[7] = 0;

  const i32x4 z4 = {0, 0, 0, 0};
#if __has_include(<hip/amd_detail/amd_gfx1250_TDM.h>)
  const i32x8 z8 = {0, 0, 0, 0, 0, 0, 0, 0};
  __builtin_amdgcn_tensor_load_to_lds(g0, g1, z4, z4, z8, 0);   // clang-23 6-arg form
#else
  __builtin_amdgcn_tensor_load_to_lds(g0, g1, z4, z4, 0);       // ROCm clang-22 5-arg form
#endif
}
#else
#define USE_TDM 0
#endif

__global__ __launch_bounds__(256)
void stgraph_gram_kernel(const float* __restrict__ spatial,
                         const float* __restrict__ temporal,
                         float* __restrict__ out) {
  __shared__ float bufA[64 * LDS_STRIDE];
  __shared__ float bufB[64 * LDS_STRIDE];

  const int bx = blockIdx.x;   // n macro-tile (64 output cols)
  const int by = blockIdx.y;   // m macro-tile (64 output rows)
  const int b  = blockIdx.z;   // batch

  const int tid  = threadIdx.x;
  const int wave = tid >> 5;
  const int lane = tid & 31;
  const int l16  = lane & 15;
  const int hi   = lane >> 4;  // K-half selector for f32 WMMA fragments

#if USE_TDM
  // 2048 is 64-aligned, so each macro-tile comes entirely from one tensor:
  // the tile base pointer is block-uniform -> scalar D# in SGPRs.
  if (wave == 0) {
    const int rowA = by * 64;
    const float* srcA = (rowA < NSPAT)
        ? spatial  + ((size_t)b * NSPAT + rowA) * DDIM
        : temporal + ((size_t)b * NTEMP + (rowA - NSPAT)) * DDIM;
    const int rowB = bx * 64;
    const float* srcB = (rowB < NSPAT)
        ? spatial  + ((size_t)b * NSPAT + rowB) * DDIM
        : temporal + ((size_t)b * NTEMP + (rowB - NSPAT)) * DDIM;
    tdm_load_tile_64x64(srcA, bufA);
    tdm_load_tile_64x64(srcB, bufB);
    __builtin_amdgcn_s_wait_tensorcnt(0);   // data landed in LDS
  }
  __syncthreads();                          // publish LDS to all 8 waves
#else
  // Manual fallback: coalesced float4 loads + ds_store, 64 rows x 16 float4s.
#pragma unroll
  for (int i = 0; i < 4; ++i) {
    const int f   = tid + i * 256;
    const int row = f >> 4;
    const int c   = (f & 15) << 2;
    {
      const int grow = by * 64 + row;
      const float* src = (grow < NSPAT)
        ? spatial  + ((size_t)b * NSPAT + grow) * DDIM + c
        : temporal + ((size_t)b * NTEMP + (grow - NSPAT)) * DDIM + c;
      *(v4f*)&bufA[row * LDS_STRIDE + c] = *(const v4f*)src;
    }
    {
      const int grow = bx * 64 + row;
      const float* src = (grow < NSPAT)
        ? spatial  + ((size_t)b * NSPAT + grow) * DDIM + c
        : temporal + ((size_t)b * NTEMP + (grow - NSPAT)) * DDIM + c;
      *(v4f*)&bufB[row * LDS_STRIDE + c] = *(const v4f*)src;
    }
  }
  __syncthreads();
#endif

  // ---- wave -> tile mapping: 8 waves cover the 4x4 grid of 16x16 tiles,
  // each wave owns two m-tiles sharing one B fragment per k-step.
  const int tn  = wave & 3;
  const int tm0 = (wave >> 2) * 2;
  const int tm1 = tm0 + 1;

  v8f acc0 = {};
  v8f acc1 = {};

  const int browB = l16 * LDS_STRIDE;                  // B frag: row n = l16
  const int arow0 = (tm0 * 16 + l16) * LDS_STRIDE;     // A frag: row m = l16
  const int arow1 = (tm1 * 16 + l16) * LDS_STRIDE;

  // K = 64 -> 16 steps of V_WMMA_F32_16X16X4_F32, 2 tiles per step.
#pragma unroll
  for (int k0 = 0; k0 < DDIM; k0 += 4) {
    const int koff = k0 + 2 * hi;                      // lane<16: K{0,1}; lane>=16: K{2,3}
    v2f bf = *(const v2f*)&bufB[browB + koff];
    v2f a0 = *(const v2f*)&bufA[arow0 + koff];
    v2f a1 = *(const v2f*)&bufA[arow1 + koff];
    acc0 = __builtin_amdgcn_wmma_f32_16x16x4_f32(false, a0, false, bf,
                                                 (short)0, acc0, false, false);
    acc1 = __builtin_amdgcn_wmma_f32_16x16x4_f32(false, a1, false, bf,
                                                 (short)0, acc1, false, false);
  }

  // ---- epilogue: tanh(relu(x)) + 0.5 on the diagonal; NT streaming stores ----
  // C/D layout: VGPR i -> row (i + 8*hi), lane%16 -> column.
  const int gn = bx * 64 + tn * 16 + l16;
  const size_t outBase = (size_t)b * MTOT * MTOT + gn;

  {
    const int gm0 = by * 64 + tm0 * 16 + hi * 8;
#pragma unroll
    for (int i = 0; i < 8; ++i) {
      const int gm = gm0 + i;
      float x = fast_tanh(fmaxf(acc0[i], 0.0f));
      if (gm == gn) x += 0.5f;
      __builtin_nontemporal_store(x, &out[outBase + (size_t)gm * MTOT]);
    }
  }
  {
    const int gm1 = by * 64 + tm1 * 16 + hi * 8;
#pragma unroll
    for (int i = 0; i < 8; ++i) {
      const int gm = gm1 + i;
      float x = fast_tanh(fmaxf(acc1[i], 0.0f));
      if (gm == gn) x += 0.5f;
      __builtin_nontemporal_store(x, &out[outBase + (size_t)gm * MTOT]);
    }
  }
}

extern "C" void kernel_launch(void* const* d_in, const int* in_sizes, int n_in,
                              void* d_out, int out_size, void* d_ws, size_t ws_size,
                              hipStream_t stream) {
  const float* spatial  = (const float*)d_in[0];   // [16, 2048, 64] f32
  const float* temporal = (const float*)d_in[1];   // [16, 64, 64]  f32
  float* out = (float*)d_out;                      // [16, 2112, 2112] f32

  dim3 grid(MTOT / 64, MTOT / 64, 16);             // (33, 33, 16)
  dim3 block(256);                                 // 8 wave32s per block
  stgraph_gram_kernel<<<grid, block, 0, stream>>>(spatial, temporal, out);
}